// CRF_62938450756203
// MI455X (gfx1250) — compile-verified
//
#include <hip/hip_runtime.h>
#include <hip/hip_bf16.h>
#include <math.h>

// CRF NLL for B=512, L=512, C=128 on gfx1250 (MI455X).
// Forward scan done as exp-space GEMM per step using V_WMMA_F32_16X16X4_F32
// (full f32 precision; scan is latency-bound so K=4 f32 WMMA is the right op).

#define Bx 512
#define Lx 512
#define Cx 128
#define MB 16        // batch rows per workgroup (WMMA M)
#define ASTR 132     // padded LDS row stride in floats (conflict-free frag reads)

typedef float v2f __attribute__((ext_vector_type(2)));
typedef float v8f __attribute__((ext_vector_type(8)));

// ---------------------------------------------------------------- gold score
__global__ void __launch_bounds__(128)
crf_gold_score(const float* __restrict__ em, const int* __restrict__ tags,
               const float* __restrict__ mask, const float* __restrict__ T,
               float* __restrict__ scores) {
  const int b   = blockIdx.x;
  const int tid = threadIdx.x;
  __shared__ float red[128];
  __shared__ int lastS;

  // last_idx = sum(mask[b]) - 1
  float ms = 0.f;
  for (int t = tid; t < Lx; t += 128) ms += mask[b * Lx + t];
  red[tid] = ms;
  __syncthreads();
  for (int s = 64; s > 0; s >>= 1) {
    if (tid < s) red[tid] += red[tid + s];
    __syncthreads();
  }
  if (tid == 0) lastS = (int)(red[0] + 0.5f) - 1;
  __syncthreads();
  const int last = lastS;

  float acc = 0.f;
  for (int t = tid; t < Lx - 1; t += 128) {
    int tg0 = tags[b * Lx + t];
    int tg1 = tags[b * Lx + t + 1];
    if (t < last)             acc += T[tg0 * Cx + tg1];                   // transition
    if (t >= 1 && t < last)   acc += em[((size_t)b * Lx + t) * Cx + tg0]; // emission
  }
  red[tid] = acc;
  __syncthreads();
  for (int s = 64; s > 0; s >>= 1) {
    if (tid < s) red[tid] += red[tid + s];
    __syncthreads();
  }
  if (tid == 0) scores[b] = red[0];
}

// ---------------------------------------------------------------- forward scan
__global__ void __launch_bounds__(256)
crf_forward(const float* __restrict__ em, const float* __restrict__ mask,
            const float* __restrict__ T, float* __restrict__ logZ) {
  __shared__ float alpha[MB * ASTR];   // log-space alphas, 16 rows x 128 cols (padded)
  __shared__ float expA [MB * ASTR];   // exp(alpha - rowmax), WMMA-A staging
  __shared__ float pmax [MB][16];
  __shared__ float rowmax[MB];
  __shared__ float mrow [MB];
  __shared__ int   li   [MB];

  const int tid  = threadIdx.x;
  const int lane = tid & 31;
  const int w    = tid >> 5;     // wave 0..7 : owns N-tile columns [16w, 16w+16)
  const int row  = tid >> 4;     // 0..15 (expA producer mapping)
  const int sub  = tid & 15;     // 0..15
  const int b0   = blockIdx.x * MB;
  const int hi   = lane >> 4;    // lane half (WMMA K split)
  const int l15  = lane & 15;
  const int col  = w * 16 + l15; // global tag column this lane produces

  // per-row last index
  if (tid < MB) {
    float s = 0.f;
    for (int t = 0; t < Lx; ++t) s += mask[(size_t)(b0 + tid) * Lx + t];
    li[tid] = (int)(s + 0.5f) - 1;
  }

  // expT in WMMA-B register layout, resident in VGPRs for all 510 steps.
  // B is 4x16 per kb: lane L, vgpr v holds B[K = 4kb + 2*(L/16) + v][N = L%16].
  v2f bfrag[32];
#pragma unroll
  for (int kb = 0; kb < 32; ++kb) {
    int j = kb * 4 + 2 * hi;
    bfrag[kb].x = __expf(T[(size_t)j * Cx + col]);
    bfrag[kb].y = __expf(T[(size_t)(j + 1) * Cx + col]);
  }

  // alpha init at t=1: alpha0_k = T[START=0][k] + e[b,1,k]
#pragma unroll
  for (int i = 0; i < 8; ++i) {
    int c = sub * 8 + i;
    alpha[row * ASTR + c] = T[c] + em[((size_t)(b0 + row) * Lx + 1) * Cx + c];
  }
  __syncthreads();

  for (int t = 2; t < Lx; ++t) {
    // ---- phase 1: parallel row max + step mask
    float m = -3.0e38f;
#pragma unroll
    for (int i = 0; i < 8; ++i) m = fmaxf(m, alpha[row * ASTR + sub * 8 + i]);
    pmax[row][sub] = m;
    if (sub == 0)
      mrow[row] = (t == li[row]) ? 0.f : mask[(size_t)(b0 + row) * Lx + t];
    __syncthreads();
    if (sub == 0) {
      float mm = pmax[row][0];
#pragma unroll
      for (int i = 1; i < 16; ++i) mm = fmaxf(mm, pmax[row][i]);
      rowmax[row] = mm;
    }
    __syncthreads();

    // ---- phase 2: expA = exp(alpha - rowmax)
    float rm = rowmax[row];
#pragma unroll
    for (int i = 0; i < 8; ++i) {
      int c = sub * 8 + i;
      expA[row * ASTR + c] = __expf(alpha[row * ASTR + c] - rm);
    }
    __syncthreads();

    // ---- phase 3: emissions early (latency hidden under the K-loop)
    float ereg[8];
#pragma unroll
    for (int r = 0; r < 8; ++r) {
      int mr = r + 8 * hi;
      ereg[r] = em[((size_t)(b0 + mr) * Lx + t) * Cx + col];
    }

    // D = expA(16x128) @ expT(128x16) : 32 chained f32 WMMAs (K=4 each)
    v8f acc = {0.f, 0.f, 0.f, 0.f, 0.f, 0.f, 0.f, 0.f};
#pragma unroll
    for (int kb = 0; kb < 32; ++kb) {
      // A 16x4: lane L, vgpr v holds A[M = L%16][K = 4kb + 2*(L/16) + v]
      v2f a = *(const v2f*)&expA[l15 * ASTR + kb * 4 + 2 * hi];
      acc = __builtin_amdgcn_wmma_f32_16x16x4_f32(
          /*neg_a=*/false, a, /*neg_b=*/false, bfrag[kb],
          /*c_mod=*/(short)0, acc, /*reuse_a=*/false, /*reuse_b=*/false);
    }

    // ---- phase 4: new alpha = log(S) + rowmax + e ; masked blend; write back
#pragma unroll
    for (int r = 0; r < 8; ++r) {
      int mr = r + 8 * hi;  // D layout: vgpr r -> M = r + 8*(lane/16), N = lane%16
      float cand = fmaxf(__logf(acc[r]), -1.0e30f) + rowmax[mr] + ereg[r];
      float mv   = mrow[mr];
      float old  = alpha[mr * ASTR + col];
      alpha[mr * ASTR + col] = mv * cand + (1.f - mv) * old;
    }
    __syncthreads();
  }

  // termination: logZ = logsumexp_k(alpha_k + T[k][END=1])
  if (tid < MB) {
    float mm = -3.0e38f;
    for (int k = 0; k < Cx; ++k)
      mm = fmaxf(mm, alpha[tid * ASTR + k] + T[(size_t)k * Cx + 1]);
    float s = 0.f;
    for (int k = 0; k < Cx; ++k)
      s += __expf(alpha[tid * ASTR + k] + T[(size_t)k * Cx + 1] - mm);
    logZ[b0 + tid] = __logf(s) + mm;
  }
}

// ---------------------------------------------------------------- final reduce
__global__ void __launch_bounds__(256)
crf_reduce(const float* __restrict__ scores, const float* __restrict__ logZ,
           float* __restrict__ out) {
  __shared__ float red[256];
  int tid = threadIdx.x;
  float s = 0.f;
  for (int b = tid; b < Bx; b += 256) s += logZ[b] - scores[b];
  red[tid] = s;
  __syncthreads();
  for (int k = 128; k > 0; k >>= 1) {
    if (tid < k) red[tid] += red[tid + k];
    __syncthreads();
  }
  if (tid == 0) out[0] = red[0];  // loss = -sum(scores - logZ)
}

// ---------------------------------------------------------------- launch
extern "C" void kernel_launch(void* const* d_in, const int* in_sizes, int n_in,
                              void* d_out, int out_size, void* d_ws, size_t ws_size,
                              hipStream_t stream) {
  (void)in_sizes; (void)n_in; (void)out_size; (void)ws_size;
  const float* em   = (const float*)d_in[0];
  const int*   tags = (const int*)d_in[1];
  const float* mask = (const float*)d_in[2];
  const float* T    = (const float*)d_in[3];

  float* scores = (float*)d_ws;       // [Bx]
  float* logZ   = scores + Bx;        // [Bx]

  crf_gold_score<<<Bx, 128, 0, stream>>>(em, tags, mask, T, scores);
  crf_forward<<<Bx / MB, 256, 0, stream>>>(em, mask, T, logZ);
  crf_reduce<<<1, 256, 0, stream>>>(scores, logZ, (float*)d_out);
}